// LocalAggregator_79783312490962
// MI455X (gfx1250) — compile-verified
//
#include <hip/hip_runtime.h>
#include <math.h>

// Problem constants (from reference)
#define P_N 16384
#define G_N 2048
#define C_N 18
#define CP_N 32              // channels padded to two 16-wide WMMA N-tiles
#define GRID_INV 2.0f        // 1 / GRID_SIZE (0.5)
#define RAD_MULT 6.0f        // SCALE_MULT / GRID_SIZE = 3.0 / 0.5

typedef __attribute__((ext_vector_type(2))) float v2f;
typedef __attribute__((ext_vector_type(8))) float v8f;

// Packed per-gaussian record, 16 floats = 64 bytes:
// [0]=mx [1]=my [2]=mz [3]=a(c00) | [4]=b(c11) [5]=c(c22) [6]=pxy(c01) [7]=pyz(c12)
// [8]=pxz(c02) [9]=imx [10]=imy [11]=imz | [12]=rad [13..15]=pad
__global__ __launch_bounds__(256) void la_prep_kernel(
    const float* __restrict__ means3D,
    const float* __restrict__ scales,
    const float* __restrict__ cov3D,
    const float* __restrict__ pc_min,
    float* __restrict__ gp)
{
    int g = blockIdx.x * blockDim.x + threadIdx.x;
    if (g >= G_N) return;

    float q0 = pc_min[0], q1 = pc_min[1], q2 = pc_min[2];
    float mx = means3D[g * 3 + 0];
    float my = means3D[g * 3 + 1];
    float mz = means3D[g * 3 + 2];

    int imx = (int)floorf((mx - q0) * GRID_INV);
    int imy = (int)floorf((my - q1) * GRID_INV);
    int imz = (int)floorf((mz - q2) * GRID_INV);

    float smax = fmaxf(scales[g * 3 + 0], fmaxf(scales[g * 3 + 1], scales[g * 3 + 2]));
    int rad = (int)ceilf(smax * RAD_MULT);

    const float* cv = cov3D + (size_t)g * 9;
    float* o = gp + (size_t)g * 16;
    o[0]  = mx;     o[1]  = my;     o[2]  = mz;     o[3]  = cv[0];   // a
    o[4]  = cv[4];  o[5]  = cv[8];  o[6]  = cv[1];  o[7]  = cv[5];   // b,c,pxy,pyz
    o[8]  = cv[2];                                                    // pxz
    o[9]  = __int_as_float(imx);
    o[10] = __int_as_float(imy);
    o[11] = __int_as_float(imz);
    o[12] = __int_as_float(rad);
    o[13] = 0.0f; o[14] = 0.0f; o[15] = 0.0f;
}

// Repack opacities: zero-pad 18 -> 32 channels and interleave gaussian K-pairs
// so a lane's B operand (both K values for one channel) is a single b64 load.
// Layout: opacP[pair q][channel n][k in pair] ; q = g>>1, element n*2 + (g&1).
__global__ __launch_bounds__(256) void la_opac_kernel(
    const float* __restrict__ opac,
    float* __restrict__ opacP)
{
    int idx = blockIdx.x * blockDim.x + threadIdx.x;   // 0 .. G_N*CP_N-1
    int g = idx >> 5;        // gaussian
    int n = idx & 31;        // padded channel
    float v = (n < C_N) ? opac[g * C_N + n] : 0.0f;
    opacP[(size_t)(g >> 1) * (2 * CP_N) + n * 2 + (g & 1)] = v;
}

// Branchless gaussian weight: no EXEC divergence, pure VALU selects.
__device__ __forceinline__ float la_weight(
    const float4 c0, const float4 c1, const float4 c2, const float4 c3,
    float px, float py, float pz, int pix, int piy, int piz)
{
    float dx = px - c0.x;
    float dy = py - c0.y;
    float dz = pz - c0.z;
    float a   = c0.w, b   = c1.x, c   = c1.y;
    float pxy = c1.z, pyz = c1.w, pxz = c2.x;
    float power = -0.5f * (a * dx * dx + b * dy * dy + c * dz * dz)
                  - (pxy * dx * dy + pyz * dy * dz + pxz * dx * dz);
    int imx = __float_as_int(c2.y);
    int imy = __float_as_int(c2.z);
    int imz = __float_as_int(c2.w);
    int rad = __float_as_int(c3.x);
    int ddx = abs(pix - imx);
    int ddy = abs(piy - imy);
    int ddz = abs(piz - imz);
    int dmax = max(ddx, max(ddy, ddz));
    // bitwise & : no short-circuit branches
    int ok = (int)(dmax <= rad) & (int)(power <= 0.0f);
    float w = __expf(power);
    return ok ? w : 0.0f;
}

// One wave32 per TWO 16-point tiles (32 points): gaussian records and opacity
// B-operands are loaded once per K-step and reused by both M-tiles, doubling
// per-wave ILP (8 WMMAs per unrolled iteration) at ~1 wave/SIMD occupancy.
// Full K sweep over 2048 gaussians with V_WMMA_F32_16X16X4_F32; two N-tiles
// cover the 18 (padded to 32) output channels.
__global__ __launch_bounds__(256) void la_main_kernel(
    const float* __restrict__ pts,
    const float* __restrict__ opacP,
    const float* __restrict__ gp,
    const float* __restrict__ pc_min,
    float* __restrict__ out)
{
    const int lane = threadIdx.x & 31;
    const int wave = threadIdx.x >> 5;
    const int half = lane >> 4;      // which 16-lane half
    const int lm   = lane & 15;

    const int p_base = (blockIdx.x * 8 + wave) * 32;
    const int pA = p_base + lm;          // M-tile 0 (A layout mirrors per half)
    const int pB = p_base + 16 + lm;     // M-tile 1

    const float q0 = pc_min[0], q1 = pc_min[1], q2 = pc_min[2];
    const float pxA = pts[pA * 3 + 0];
    const float pyA = pts[pA * 3 + 1];
    const float pzA = pts[pA * 3 + 2];
    const float pxB = pts[pB * 3 + 0];
    const float pyB = pts[pB * 3 + 1];
    const float pzB = pts[pB * 3 + 2];
    const int pixA = (int)floorf((pxA - q0) * GRID_INV);
    const int piyA = (int)floorf((pyA - q1) * GRID_INV);
    const int pizA = (int)floorf((pzA - q2) * GRID_INV);
    const int pixB = (int)floorf((pxB - q0) * GRID_INV);
    const int piyB = (int)floorf((pyB - q1) * GRID_INV);
    const int pizB = (int)floorf((pzB - q2) * GRID_INV);

    const float4* __restrict__ gp4 = (const float4*)gp;
    const float2* __restrict__ opacP2 = (const float2*)opacP;

    v8f accA0 = {0.f, 0.f, 0.f, 0.f, 0.f, 0.f, 0.f, 0.f};  // tile0, ch 0..15
    v8f accA1 = {0.f, 0.f, 0.f, 0.f, 0.f, 0.f, 0.f, 0.f};  // tile0, ch 16..31
    v8f accB0 = {0.f, 0.f, 0.f, 0.f, 0.f, 0.f, 0.f, 0.f};  // tile1, ch 0..15
    v8f accB1 = {0.f, 0.f, 0.f, 0.f, 0.f, 0.f, 0.f, 0.f};  // tile1, ch 16..31

#pragma unroll 2
    for (int g0 = 0; g0 < G_N; g0 += 4) {
        const int ga = g0 + 2 * half;   // this lane's K pair: ga, ga+1 (ga even)
        const int qp = ga >> 1;         // pair index

        // gaussian ga
        float4 a0c0 = gp4[ga * 4 + 0];
        float4 a0c1 = gp4[ga * 4 + 1];
        float4 a0c2 = gp4[ga * 4 + 2];
        float4 a0c3 = gp4[ga * 4 + 3];
        // gaussian ga+1
        float4 a1c0 = gp4[ga * 4 + 4];
        float4 a1c1 = gp4[ga * 4 + 5];
        float4 a1c2 = gp4[ga * 4 + 6];
        float4 a1c3 = gp4[ga * 4 + 7];

        // B operands: one b64 each, unconditional (zero-padded channels),
        // shared by both M-tiles.
        float2 b0 = opacP2[qp * CP_N + lm];
        float2 b1 = opacP2[qp * CP_N + 16 + lm];
        v2f B0; B0.x = b0.x; B0.y = b0.y;
        v2f B1; B1.x = b1.x; B1.y = b1.y;

        v2f Aa, Ab;
        Aa.x = la_weight(a0c0, a0c1, a0c2, a0c3, pxA, pyA, pzA, pixA, piyA, pizA);
        Aa.y = la_weight(a1c0, a1c1, a1c2, a1c3, pxA, pyA, pzA, pixA, piyA, pizA);
        Ab.x = la_weight(a0c0, a0c1, a0c2, a0c3, pxB, pyB, pzB, pixB, piyB, pizB);
        Ab.y = la_weight(a1c0, a1c1, a1c2, a1c3, pxB, pyB, pzB, pixB, piyB, pizB);

        accA0 = __builtin_amdgcn_wmma_f32_16x16x4_f32(
            false, Aa, false, B0, (short)0, accA0, false, false);
        accA1 = __builtin_amdgcn_wmma_f32_16x16x4_f32(
            false, Aa, false, B1, (short)0, accA1, false, false);
        accB0 = __builtin_amdgcn_wmma_f32_16x16x4_f32(
            false, Ab, false, B0, (short)0, accB0, false, false);
        accB1 = __builtin_amdgcn_wmma_f32_16x16x4_f32(
            false, Ab, false, B1, (short)0, accB1, false, false);
    }

    // D layout: lane l, reg r -> (M = r + 8*half, N = l&15)
#pragma unroll
    for (int r = 0; r < 8; ++r) {
        const int rowA = p_base + r + 8 * half;
        const int rowB = p_base + 16 + r + 8 * half;
        out[rowA * C_N + lm] = accA0[r];
        out[rowB * C_N + lm] = accB0[r];
        if (lm < 2) {
            out[rowA * C_N + 16 + lm] = accA1[r];
            out[rowB * C_N + 16 + lm] = accB1[r];
        }
    }
}

extern "C" void kernel_launch(void* const* d_in, const int* in_sizes, int n_in,
                              void* d_out, int out_size, void* d_ws, size_t ws_size,
                              hipStream_t stream) {
    const float* pts     = (const float*)d_in[0];
    const float* means3D = (const float*)d_in[1];
    const float* opac    = (const float*)d_in[2];
    const float* scales  = (const float*)d_in[3];
    const float* cov3D   = (const float*)d_in[4];
    const float* pc_min  = (const float*)d_in[5];
    float* out = (float*)d_out;

    float* gp    = (float*)d_ws;                       // 2048*64B  = 128 KB
    float* opacP = gp + (size_t)G_N * 16;              // 2048*32*4 = 256 KB

    la_prep_kernel<<<(G_N + 255) / 256, 256, 0, stream>>>(means3D, scales, cov3D, pc_min, gp);
    la_opac_kernel<<<(G_N * CP_N) / 256, 256, 0, stream>>>(opac, opacP);
    la_main_kernel<<<P_N / 256, 256, 0, stream>>>(pts, opacP, gp, pc_min, out);
}